// update_v_60653528154555
// MI455X (gfx1250) — compile-verified
//
#include <hip/hip_runtime.h>

#define N_NODES 50000
#define N_EDGES 800000
#define HIDDEN  128
#define LDS_STRIDE 130  // pad 128->130 floats: rows land on distinct banks

typedef __attribute__((ext_vector_type(2))) float v2f;
typedef __attribute__((ext_vector_type(8))) float v8f;

// ---------------------------------------------------------------------------
// Kernel 1: zero the aggregation workspace (harness poisons ws with 0xAA).
// ---------------------------------------------------------------------------
__global__ void zero_kernel(float4* __restrict__ p, int n4) {
    int i = blockIdx.x * blockDim.x + threadIdx.x;
    int stride = gridDim.x * blockDim.x;
    float4 z = make_float4(0.f, 0.f, 0.f, 0.f);
    for (; i < n4; i += stride) p[i] = z;
}

// ---------------------------------------------------------------------------
// Kernel 2: scatter-sum edge features into agg[dst].
// 256 threads = 2 edges/block, 128 lanes per edge (coalesced row access).
// ---------------------------------------------------------------------------
__global__ void scatter_kernel(const float* __restrict__ e,
                               const int*   __restrict__ edge_index,
                               float* __restrict__ agg) {
    int edge = blockIdx.x * 2 + (threadIdx.x >> 7);
    int feat = threadIdx.x & 127;
    if (edge < N_EDGES) {
        int dst = edge_index[N_EDGES + edge];  // row 1 of [2, N_EDGES]
        atomicAdd(&agg[(size_t)dst * HIDDEN + feat],
                  e[(size_t)edge * HIDDEN + feat]);
    }
}

// ---------------------------------------------------------------------------
// Kernel 3: out = v + relu(agg@W1 + b1)@W2 + b2, 16-node tile per block.
// 8 waves; wave w owns output columns [16w, 16w+16). fp32 WMMA 16x16x4.
// ---------------------------------------------------------------------------
__launch_bounds__(256, 2)
__global__ void mlp_kernel(const float* __restrict__ agg,
                           const float* __restrict__ v,
                           const float* __restrict__ W1,
                           const float* __restrict__ b1,
                           const float* __restrict__ W2,
                           const float* __restrict__ b2,
                           float* __restrict__ out) {
    __shared__ float lds_a[16 * LDS_STRIDE];
    __shared__ float lds_h[16 * LDS_STRIDE];

    const int tid     = threadIdx.x;
    const int wave    = tid >> 5;
    const int lane    = tid & 31;
    const int half    = lane >> 4;     // 0: lanes 0-15, 1: lanes 16-31
    const int l16     = lane & 15;
    const int colbase = wave * 16;
    const int node0   = blockIdx.x * 16;

    // Stage the contiguous 16x128 agg tile into LDS (padded stride).
    const float* atile = agg + (size_t)node0 * HIDDEN;
    for (int i = tid; i < 16 * HIDDEN; i += 256) {
        int r = i >> 7, c = i & 127;
        lds_a[r * LDS_STRIDE + c] = atile[i];
    }
    __syncthreads();

    // ---- GEMM1: h = relu(A @ W1 + b1) ------------------------------------
    // A frag (16x4 f32): lane<16 -> row=lane, K=k0..k0+1; lane>=16 -> K=k0+2..3
    // B frag (4x16 f32): lane<16 -> col=lane, K=k0..k0+1 in .x/.y; etc.
    v8f c1 = {0.f, 0.f, 0.f, 0.f, 0.f, 0.f, 0.f, 0.f};
#pragma unroll
    for (int kk = 0; kk < HIDDEN / 4; ++kk) {
        int k = kk * 4 + half * 2;
        v2f a, b;
        a.x = lds_a[l16 * LDS_STRIDE + k];
        a.y = lds_a[l16 * LDS_STRIDE + k + 1];
        b.x = W1[(size_t)k       * HIDDEN + colbase + l16];
        b.y = W1[(size_t)(k + 1) * HIDDEN + colbase + l16];
        c1 = __builtin_amdgcn_wmma_f32_16x16x4_f32(
                 false, a, false, b, (short)0, c1, false, false);
    }

    // C layout: VGPR g holds row M = g + half*8, col N = colbase + l16.
    float bias1 = b1[colbase + l16];
#pragma unroll
    for (int g = 0; g < 8; ++g) {
        int m = g + half * 8;
        float hv = c1[g] + bias1;
        hv = hv > 0.f ? hv : 0.f;
        lds_h[m * LDS_STRIDE + colbase + l16] = hv;
    }
    __syncthreads();

    // ---- GEMM2: out = h @ W2 + b2 + v ------------------------------------
    v8f c2 = {0.f, 0.f, 0.f, 0.f, 0.f, 0.f, 0.f, 0.f};
#pragma unroll
    for (int kk = 0; kk < HIDDEN / 4; ++kk) {
        int k = kk * 4 + half * 2;
        v2f a, b;
        a.x = lds_h[l16 * LDS_STRIDE + k];
        a.y = lds_h[l16 * LDS_STRIDE + k + 1];
        b.x = W2[(size_t)k       * HIDDEN + colbase + l16];
        b.y = W2[(size_t)(k + 1) * HIDDEN + colbase + l16];
        c2 = __builtin_amdgcn_wmma_f32_16x16x4_f32(
                 false, a, false, b, (short)0, c2, false, false);
    }

    float bias2 = b2[colbase + l16];
    const float* vtile = v + (size_t)node0 * HIDDEN;
#pragma unroll
    for (int g = 0; g < 8; ++g) {
        int m = g + half * 8;
        out[(size_t)(node0 + m) * HIDDEN + colbase + l16] =
            c2[g] + bias2 + vtile[m * HIDDEN + colbase + l16];
    }
}

// ---------------------------------------------------------------------------
extern "C" void kernel_launch(void* const* d_in, const int* in_sizes, int n_in,
                              void* d_out, int out_size, void* d_ws, size_t ws_size,
                              hipStream_t stream) {
    const float* v  = (const float*)d_in[0];
    const float* e  = (const float*)d_in[1];
    const int*   ei = (const int*)d_in[2];   // [2, N_EDGES], int
    const float* W1 = (const float*)d_in[3];
    const float* b1 = (const float*)d_in[4];
    const float* W2 = (const float*)d_in[5];
    const float* b2 = (const float*)d_in[6];
    float* out = (float*)d_out;
    float* agg = (float*)d_ws;               // 50000*128 floats = 25.6 MB

    zero_kernel<<<2048, 256, 0, stream>>>((float4*)agg, N_NODES * HIDDEN / 4);
    scatter_kernel<<<N_EDGES / 2, 256, 0, stream>>>(e, ei, agg);
    mlp_kernel<<<N_NODES / 16, 256, 0, stream>>>(agg, v, W1, b1, W2, b2, out);
}